// GCN_29214367547480
// MI455X (gfx1250) — compile-verified
//
#include <hip/hip_runtime.h>
#include <hip/hip_bf16.h>

typedef __attribute__((ext_vector_type(2))) float v2f;
typedef __attribute__((ext_vector_type(8))) float v8f;

#define ODIM 64

// ---------------------------------------------------------------------------
// Dense GEMM with CDNA5 WMMA: C[M x 64] = A[M x K] @ W[K x 64], fp32.
// One wave per 16x16 output tile; 4 waves per block cover the full 64 cols.
// Uses V_WMMA_F32_16X16X4_F32 (wave32, 8 accumulator VGPRs).
// ---------------------------------------------------------------------------
__global__ __launch_bounds__(128) void gemm_wmma_f32(
    const float* __restrict__ A, const float* __restrict__ W,
    float* __restrict__ C, int M, int K) {
  const int lane = threadIdx.x & 31;
  const int wave = threadIdx.x >> 5;     // 0..3 -> N tile
  const int nb   = wave << 4;            // column base
  const int half = lane >> 4;            // 0: lanes 0-15, 1: lanes 16-31
  const int l15  = lane & 15;
  const int mtiles = M >> 4;

  for (int mt = blockIdx.x; mt < mtiles; mt += gridDim.x) {
    const int mbase = mt << 4;
    v8f acc = {};
    const float* __restrict__ arow = A + (size_t)(mbase + l15) * K;
    for (int k = 0; k < K; k += 4) {
      // A fragment (16x4): lanes 0-15 carry K = k,k+1 ; lanes 16-31 carry K = k+2,k+3
      v2f a;
      a[0] = arow[k + 2 * half + 0];
      a[1] = arow[k + 2 * half + 1];
      // B fragment (4x16): row K striped across lanes within a VGPR
      v2f b;
      b[0] = W[(size_t)(k + 2 * half + 0) * ODIM + nb + l15];
      b[1] = W[(size_t)(k + 2 * half + 1) * ODIM + nb + l15];
      acc = __builtin_amdgcn_wmma_f32_16x16x4_f32(
          /*neg_a=*/false, a, /*neg_b=*/false, b,
          /*c_mod=*/(short)0, acc, /*reuse_a=*/false, /*reuse_b=*/false);
    }
    // D layout: VGPR r -> row mbase + r + 8*half, col nb + l15
    float* __restrict__ crow = C + (size_t)(mbase + 8 * half) * ODIM + nb + l15;
#pragma unroll
    for (int r = 0; r < 8; ++r) crow[(size_t)r * ODIM] = acc[r];
  }
}

// ---------------------------------------------------------------------------
// Degree / normalization
// ---------------------------------------------------------------------------
__global__ void init_deg(float* __restrict__ deg, int N) {
  int i = blockIdx.x * blockDim.x + threadIdx.x;
  if (i < N) deg[i] = 1.0f;  // self-loop weight
}

__global__ void edge_deg(const int* __restrict__ dst,
                         const float* __restrict__ w,
                         float* __restrict__ deg, int E) {
  int e = blockIdx.x * blockDim.x + threadIdx.x;
  if (e < E) atomicAdd(&deg[dst[e]], w[e]);
}

__global__ void make_dinv(const float* __restrict__ deg,
                          float* __restrict__ dinv, int N) {
  int i = blockIdx.x * blockDim.x + threadIdx.x;
  if (i < N) {
    float d = deg[i];
    dinv[i] = d > 0.0f ? __frsqrt_rn(d) : 0.0f;
  }
}

// ---------------------------------------------------------------------------
// h[i,:] = b + dinv[i]^2 * xw[i,:]   (fused bias + self-loop contribution)
// ---------------------------------------------------------------------------
__global__ __launch_bounds__(256) void init_h(
    const float* __restrict__ xw, const float* __restrict__ dinv,
    const float* __restrict__ b, float* __restrict__ h, int N) {
  long long t = (long long)blockIdx.x * blockDim.x + threadIdx.x;
  int i = (int)(t >> 6);
  int f = (int)(t & 63);
  if (i < N) {
    float di = dinv[i];
    h[(size_t)i * ODIM + f] = b[f] + di * di * xw[(size_t)i * ODIM + f];
  }
}

// ---------------------------------------------------------------------------
// Edge aggregation: h[dst,:] += dinv[src]*w*dinv[dst] * xw[src,:]
// 64 consecutive threads per edge -> coalesced gather + coalesced f32 atomics,
// all hitting the 192MB L2 (feature matrix is 25.6MB).
// ---------------------------------------------------------------------------
__global__ __launch_bounds__(256) void gcn_aggregate(
    const float* __restrict__ xw, const float* __restrict__ dinv,
    const int* __restrict__ src, const int* __restrict__ dst,
    const float* __restrict__ w, float* __restrict__ h, int E) {
  long long t = (long long)blockIdx.x * blockDim.x + threadIdx.x;
  long long e = t >> 6;
  int f = (int)(t & 63);
  if (e >= E) return;
  int s = src[e];
  int d = dst[e];
  float norm = dinv[s] * w[e] * dinv[d];
  atomicAdd(&h[(size_t)d * ODIM + f], norm * xw[(size_t)s * ODIM + f]);
}

// ---------------------------------------------------------------------------
// Pooling: pairwise feature max + min batch id per cluster
// ---------------------------------------------------------------------------
__global__ __launch_bounds__(256) void pool_max(
    const float* __restrict__ h, const int* __restrict__ batch,
    float* __restrict__ xp, int* __restrict__ bp, int nc) {
  long long t = (long long)blockIdx.x * blockDim.x + threadIdx.x;
  int c = (int)(t >> 6);
  int f = (int)(t & 63);
  if (c >= nc) return;
  size_t r0 = (size_t)(2 * c) * ODIM;
  xp[(size_t)c * ODIM + f] = fmaxf(h[r0 + f], h[r0 + ODIM + f]);
  if (f == 0) bp[c] = min(batch[2 * c], batch[2 * c + 1]);
}

__global__ void init_first(int* __restrict__ first, int B) {
  int g = blockIdx.x * blockDim.x + threadIdx.x;
  if (g < B) first[g] = 0x7FFFFFFF;
}

__global__ void scatter_first(const int* __restrict__ bp,
                              int* __restrict__ first, int nc) {
  int c = blockIdx.x * blockDim.x + threadIdx.x;
  if (c < nc) atomicMin(&first[bp[c]], c);
}

__global__ void gather_out(const float* __restrict__ xp,
                           const int* __restrict__ first,
                           float* __restrict__ out, int B, int nc) {
  long long t = (long long)blockIdx.x * blockDim.x + threadIdx.x;
  int g = (int)(t >> 6);
  int f = (int)(t & 63);
  if (g >= B) return;
  int c = first[g];
  if (c > nc - 1) c = nc - 1;   // matches JAX clipped OOB gather for empty graphs
  if (c < 0) c = 0;
  out[(size_t)g * ODIM + f] = xp[(size_t)c * ODIM + f];
}

// ---------------------------------------------------------------------------
// Launch
// ---------------------------------------------------------------------------
extern "C" void kernel_launch(void* const* d_in, const int* in_sizes, int n_in,
                              void* d_out, int out_size, void* d_ws, size_t ws_size,
                              hipStream_t stream) {
  const float* x   = (const float*)d_in[0];     // N x 128
  const int*   ei  = (const int*)d_in[1];       // 2 x E
  const float* ew  = (const float*)d_in[2];     // E
  const int*   bat = (const int*)d_in[3];       // N
  const float* W1  = (const float*)d_in[4];     // 128 x 64
  const float* b1  = (const float*)d_in[5];     // 64
  const float* W2  = (const float*)d_in[6];     // 64 x 64
  const float* b2  = (const float*)d_in[7];     // 64
  float* out = (float*)d_out;

  const int IN_DIM = 128;
  const int N  = in_sizes[0] / IN_DIM;          // 100000
  const int E  = in_sizes[2];                   // 1600000
  const int B  = out_size / ODIM;               // 256
  const int NC = N / 2;

  const int* src = ei;
  const int* dst = ei + E;

  // workspace carve-up (all offsets 16B-aligned)
  char* ws = (char*)d_ws;
  size_t featBytes = (size_t)N * ODIM * sizeof(float);   // 25.6 MB
  float* xw   = (float*)(ws);                            // GEMM output (both layers)
  float* h    = (float*)(ws + featBytes);                // aggregated output
  float* deg  = (float*)(ws + 2 * featBytes);
  float* dinv = (float*)(ws + 2 * featBytes + (size_t)N * 4);
  float* xp   = (float*)(ws + 2 * featBytes + (size_t)N * 8);
  int*   bp   = (int*)  (ws + 2 * featBytes + (size_t)N * 8 + (size_t)NC * ODIM * 4);
  int*   first= (int*)  (ws + 2 * featBytes + (size_t)N * 8 + (size_t)NC * ODIM * 4 + (size_t)NC * 4);

  const int T = 256;
  dim3 blk(T);

  // normalization (shared by both layers)
  init_deg<<<(N + T - 1) / T, blk, 0, stream>>>(deg, N);
  edge_deg<<<(E + T - 1) / T, blk, 0, stream>>>(dst, ew, deg, E);
  make_dinv<<<(N + T - 1) / T, blk, 0, stream>>>(deg, dinv, N);

  long long nfeat = (long long)N * ODIM;
  long long efeat = (long long)E * ODIM;
  int nfeatBlocks = (int)((nfeat + T - 1) / T);
  int efeatBlocks = (int)((efeat + T - 1) / T);

  // ---- layer 1: xw = x @ W1 ; h = b1 + selfloop + scatter ----
  gemm_wmma_f32<<<N / 16, dim3(128), 0, stream>>>(x, W1, xw, N, IN_DIM);
  init_h<<<nfeatBlocks, blk, 0, stream>>>(xw, dinv, b1, h, N);
  gcn_aggregate<<<efeatBlocks, blk, 0, stream>>>(xw, dinv, src, dst, ew, h, E);

  // ---- layer 2: xw = h @ W2 ; h = b2 + selfloop + scatter ----
  gemm_wmma_f32<<<N / 16, dim3(128), 0, stream>>>(h, W2, xw, N, ODIM);
  init_h<<<nfeatBlocks, blk, 0, stream>>>(xw, dinv, b2, h, N);
  gcn_aggregate<<<efeatBlocks, blk, 0, stream>>>(xw, dinv, src, dst, ew, h, E);

  // ---- pooling ----
  long long cfeat = (long long)NC * ODIM;
  pool_max<<<(int)((cfeat + T - 1) / T), blk, 0, stream>>>(h, bat, xp, bp, NC);
  init_first<<<(B + T - 1) / T, blk, 0, stream>>>(first, B);
  scatter_first<<<(NC + T - 1) / T, blk, 0, stream>>>(bp, first, NC);
  gather_out<<<(B * ODIM + T - 1) / T, blk, 0, stream>>>(xp, first, out, B, NC);
}